// Decoder_86182813762330
// MI455X (gfx1250) — compile-verified
//
#include <hip/hip_runtime.h>
#include <hip/hip_bf16.h>
#include <math.h>

typedef __attribute__((ext_vector_type(2))) float v2f;
typedef __attribute__((ext_vector_type(8))) float v8f;

// ---------------------------------------------------------------------------
// Generic fp32 WMMA GEMM:  C[M,N] = A[M,K] @ B[K,N] (+ bias[N])
// Row-major everywhere. One wave computes a (16*MT) x 16 output strip using
// V_WMMA_F32_16X16X4_F32, stacking MT accumulators in M so each B fragment is
// reused MT times. Requires M % (16*MT) == 0, N % 16 == 0, K % 4 == 0.
// ---------------------------------------------------------------------------
template <int MT>
__global__ __launch_bounds__(256) void wmma_gemm_f32(
    const float* __restrict__ A, const float* __restrict__ B,
    const float* __restrict__ bias, float* __restrict__ C,
    int M, int N, int K)
{
    const int wavesPerBlock = blockDim.x >> 5;
    const int wid    = blockIdx.x * wavesPerBlock + (threadIdx.x >> 5);
    const int nTiles = N >> 4;
    const int mBlks  = M / (16 * MT);
    const int wm = wid / nTiles;
    const int wn = wid % nTiles;
    if (wm >= mBlks) return;                 // wave-uniform: EXEC stays all-1s

    const int lane = threadIdx.x & 31;
    const int half = lane >> 4;              // 0: lanes 0-15, 1: lanes 16-31
    const int lrow = lane & 15;
    const int col  = (wn << 4) + lrow;
    const int rowBase = wm * (16 * MT);

    v8f acc[MT];
#pragma unroll
    for (int i = 0; i < MT; ++i) acc[i] = (v8f){0,0,0,0,0,0,0,0};

    const float* Bp = B + col;
    for (int kb = 0; kb < K; kb += 4) {
        // B fragment (4x16): VGPR0 = rows K{0,2}, VGPR1 = rows K{1,3}
        const int k0 = kb + (half << 1);
        v2f bfrag;
        bfrag.x = Bp[(size_t)k0 * N];
        bfrag.y = Bp[(size_t)(k0 + 1) * N];
#pragma unroll
        for (int mt = 0; mt < MT; ++mt) {
            // A fragment (16x4): lane holds K = {k0, k0+1} of its row (contiguous)
            const float* Ap = A + (size_t)(rowBase + mt * 16 + lrow) * K + k0;
            v2f afrag = *(const v2f*)Ap;
            acc[mt] = __builtin_amdgcn_wmma_f32_16x16x4_f32(
                false, afrag, false, bfrag, (short)0, acc[mt], false, false);
        }
    }

    const float bv = bias ? bias[col] : 0.0f;
#pragma unroll
    for (int mt = 0; mt < MT; ++mt) {
#pragma unroll
        for (int r = 0; r < 8; ++r) {
            // C/D layout: VGPR r -> row r (lanes 0-15) / row r+8 (lanes 16-31)
            const int row = rowBase + mt * 16 + (half << 3) + r;
            C[(size_t)row * N + col] = acc[mt][r] + bv;
        }
    }
}

// ---------------------------------------------------------------------------
// score[b,s] = Vb + sum_u tanh(q[b,u] + K[b*S+s,u]) * Vw[u]     (one wave/row)
// ---------------------------------------------------------------------------
__global__ __launch_bounds__(256) void score_kernel(
    const float* __restrict__ Kmat, const float* __restrict__ q,
    const float* __restrict__ Vw, const float* __restrict__ Vb,
    float* __restrict__ score, int rows)
{
    const int wid = (blockIdx.x * blockDim.x + threadIdx.x) >> 5;
    if (wid >= rows) return;
    const int lane = threadIdx.x & 31;
    const int b = wid >> 6;                       // S == 64
    const float* krow = Kmat + (size_t)wid * 1024;
    const float* qrow = q + (size_t)b * 1024;
    float s = 0.0f;
    for (int u = lane; u < 1024; u += 32)
        s += tanhf(qrow[u] + krow[u]) * Vw[u];
    for (int off = 16; off > 0; off >>= 1)
        s += __shfl_down(s, off, 32);
    if (lane == 0) score[wid] = s + Vb[0];
}

// ---------------------------------------------------------------------------
// Per batch-row: softmax over S, attn out, context = attn·enc, gin = [ctx|emb]
// ---------------------------------------------------------------------------
__global__ __launch_bounds__(256) void softmax_context_kernel(
    const float* __restrict__ score, const float* __restrict__ enc,
    const float* __restrict__ emb, const int* __restrict__ x,
    float* __restrict__ attn_out, float* __restrict__ gin)
{
    __shared__ float sattn[64];
    const int b = blockIdx.x;
    const int tid = threadIdx.x;

    if (tid < 64) sattn[tid] = score[b * 64 + tid];
    __syncthreads();
    if (tid == 0) {
        float m = sattn[0];
        for (int i = 1; i < 64; ++i) m = fmaxf(m, sattn[i]);
        float sum = 0.0f;
        for (int i = 0; i < 64; ++i) { float e = expf(sattn[i] - m); sattn[i] = e; sum += e; }
        const float inv = 1.0f / sum;
        for (int i = 0; i < 64; ++i) sattn[i] *= inv;
    }
    __syncthreads();
    if (tid < 64) attn_out[b * 64 + tid] = sattn[tid];

    const float* encb = enc + (size_t)b * 64 * 1024;
    for (int u = tid; u < 1024; u += blockDim.x) {
        float c = 0.0f;
        for (int s = 0; s < 64; ++s) c += sattn[s] * encb[s * 1024 + u];
        gin[(size_t)b * 1280 + u] = c;
    }
    const int token = x[b];
    for (int e = tid; e < 256; e += blockDim.x)
        gin[(size_t)b * 1280 + 1024 + e] = emb[(size_t)token * 256 + e];
}

// ---------------------------------------------------------------------------
// GRU gates with h_prev == 0:  mh = gru_b[1];  h = (1-z) * hcand
// ---------------------------------------------------------------------------
__global__ __launch_bounds__(256) void gru_kernel(
    const float* __restrict__ mx, const float* __restrict__ gru_b,
    float* __restrict__ h_out)
{
    const int idx = blockIdx.x * blockDim.x + threadIdx.x;   // 0 .. 64*1024-1
    const int b = idx >> 10;
    const int u = idx & 1023;
    const float* m  = mx + (size_t)b * 3072;
    const float* gb = gru_b + 3072;                          // row 1
    const float xz = m[u], xr = m[1024 + u], xh = m[2048 + u];
    const float hz = gb[u], hr = gb[1024 + u], hh = gb[2048 + u];
    const float z  = 1.0f / (1.0f + expf(-(xz + hz)));
    const float r  = 1.0f / (1.0f + expf(-(xr + hr)));
    const float hc = tanhf(xh + r * hh);
    h_out[idx] = (1.0f - z) * hc;
}

// ---------------------------------------------------------------------------
extern "C" void kernel_launch(void* const* d_in, const int* in_sizes, int n_in,
                              void* d_out, int out_size, void* d_ws, size_t ws_size,
                              hipStream_t stream)
{
    const int*   x      = (const int*)  d_in[0];
    const float* dech   = (const float*)d_in[1];
    const float* enc    = (const float*)d_in[2];
    const float* W1     = (const float*)d_in[3];
    const float* b1     = (const float*)d_in[4];
    const float* W2     = (const float*)d_in[5];
    const float* b2     = (const float*)d_in[6];
    const float* Vw     = (const float*)d_in[7];
    const float* Vb     = (const float*)d_in[8];
    const float* emb    = (const float*)d_in[9];
    const float* gru_k  = (const float*)d_in[10];
    /* gru_rk (d_in[11]) unused: h_prev == 0 */
    const float* gru_b  = (const float*)d_in[12];
    const float* fc_W   = (const float*)d_in[13];
    const float* fc_b   = (const float*)d_in[14];

    float* out      = (float*)d_out;
    float* logits   = out;                       // 64*32000
    float* h_out    = out + 64 * 32000;          // 64*1024
    float* attn_out = h_out + 64 * 1024;         // 64*64

    char* ws = (char*)d_ws;
    float* q    = (float*)ws; ws += (size_t)64 * 1024 * 4;
    float* Kmat = (float*)ws; ws += (size_t)4096 * 1024 * 4;
    float* sc   = (float*)ws; ws += (size_t)4096 * 4;
    float* gin  = (float*)ws; ws += (size_t)64 * 1280 * 4;
    float* mx   = (float*)ws; ws += (size_t)64 * 3072 * 4;

    // q = dec_hidden @ W1 + b1           (64 x 1024 x 1024)  -> 64 waves
    wmma_gemm_f32<4><<<8, 256, 0, stream>>>(dech, W1, b1, q, 64, 1024, 1024);
    // K = enc @ W2 + b2                  (4096 x 1024 x 1024) -> 4096 waves
    wmma_gemm_f32<4><<<512, 256, 0, stream>>>(enc, W2, b2, Kmat, 4096, 1024, 1024);
    // score = tanh(q+K)·Vw + Vb          (one wave per (b,s) row)
    score_kernel<<<512, 256, 0, stream>>>(Kmat, q, Vw, Vb, sc, 4096);
    // softmax + context + embedding gather
    softmax_context_kernel<<<64, 256, 0, stream>>>(sc, enc, emb, x, attn_out, gin);
    // mx = gin @ gru_k + gru_b[0]        (64 x 1280 x 3072)  -> 192 waves
    wmma_gemm_f32<4><<<24, 256, 0, stream>>>(gin, gru_k, gru_b, mx, 64, 3072, 1280);
    // GRU gates -> h
    gru_kernel<<<256, 256, 0, stream>>>(mx, gru_b, h_out);
    // logits = h @ fc_W + fc_b           (64 x 1024 x 32000) -> 2000 waves
    wmma_gemm_f32<4><<<250, 256, 0, stream>>>(h_out, fc_W, fc_b, logits, 64, 32000, 1024);
}